// CoPE2d_40252433498180
// MI455X (gfx1250) — compile-verified
//
#include <hip/hip_runtime.h>
#include <hip/hip_bf16.h>

#define B_      8
#define HEADS_  12
#define HW_     24      // H == W
#define SEQ_    576     // HW_*HW_
#define HDIM_   64
#define NPOS_   24
#define BH_     (B_*HEADS_)          // 96
#define TILES_  (SEQ_/16)            // 36

typedef float v2f __attribute__((ext_vector_type(2)));
typedef float v8f __attribute__((ext_vector_type(8)));

// ---------------------------------------------------------------------------
// Stage 1: CoPE units. One wave handles one (b, head, i, branch) unit:
//   - stage q rows [24x64] and pos_emb [64x24] in LDS
//   - logits_int[24][24] = q . pos_emb  (VALU, tiny)
//   - lanes 0..23: sigmoid -> reverse cumsum -> clamp -> floor/ceil lerp
//   - emit pe point-set rows [24 floats] + squared norm per row
// branch 0 = x, branch 1 = y (selected by blockIdx.y)
// ---------------------------------------------------------------------------
__global__ __launch_bounds__(128) void cope_stage1(
    const float* __restrict__ query,   // [B,HEADS,SEQ,HDIM]
    const float* __restrict__ attn,    // [B,HEADS,SEQ,SEQ]
    const float* __restrict__ pemb_x,  // [HDIM,NPOS]
    const float* __restrict__ pemb_y,  // [HDIM,NPOS]
    float* __restrict__ pe_x,          // [BH,SEQ,NPOS]
    float* __restrict__ pe_y,          // [BH,SEQ,NPOS]
    float* __restrict__ sq_x,          // [BH,SEQ]
    float* __restrict__ sq_y)          // [BH,SEQ]
{
    __shared__ float pe_s[HDIM_ * NPOS_];        // 6 KB, shared by block
    __shared__ float q_s[4][HW_ * HDIM_];        // 4 * 6 KB
    __shared__ float li_s[4][HW_ * NPOS_];       // 4 * 2.25 KB

    const int lane = threadIdx.x & 31;
    const int wv   = threadIdx.x >> 5;
    const int branch = blockIdx.y;               // 0 = x, 1 = y
    const int u = blockIdx.x * 4 + wv;           // unit in [0, 2304)
    const int b   = u / (HEADS_ * HW_);
    const int rem = u % (HEADS_ * HW_);
    const int hd  = rem / HW_;
    const int i   = rem % HW_;
    const int bh  = b * HEADS_ + hd;

    // pos_emb -> LDS (block cooperative; branch uniform per block)
    const float* pemb = branch ? pemb_y : pemb_x;
    for (int f = threadIdx.x; f < HDIM_ * NPOS_; f += blockDim.x)
        pe_s[f] = pemb[f];

    // q slice -> LDS (wave cooperative)
    const size_t qbase = (size_t)bh * SEQ_ * HDIM_;
    if (branch == 0) {
        // rows i*24 .. i*24+23 are contiguous
        const float* src = query + qbase + (size_t)(i * HW_) * HDIM_;
        for (int f = lane; f < HW_ * HDIM_; f += 32) q_s[wv][f] = src[f];
    } else {
        for (int f = lane; f < HW_ * HDIM_; f += 32) {
            int j = f >> 6, d = f & 63;
            q_s[wv][f] = query[qbase + (size_t)(j * HW_ + i) * HDIM_ + d];
        }
    }
    __syncthreads();

    // logits_int[j][p] = sum_d q[j][d] * pe[d][p]
    for (int o = lane; o < HW_ * NPOS_; o += 32) {
        const int j = o / NPOS_, p = o % NPOS_;
        const float* qr = &q_s[wv][j * HDIM_];
        float acc = 0.f;
        #pragma unroll
        for (int d = 0; d < HDIM_; ++d)
            acc = __builtin_fmaf(qr[d], pe_s[d * NPOS_ + p], acc);
        li_s[wv][o] = acc;
    }
    __syncthreads();

    // CoPE row: lane j in [0,24)
    if (lane < HW_) {
        const int j = lane;
        const size_t abase = (size_t)bh * SEQ_ * SEQ_;
        float g[NPOS_];
        if (branch == 0) {
            const float* ar = attn + abase + (size_t)(i * HW_ + j) * SEQ_ + i * HW_;
            #pragma unroll
            for (int k = 0; k < NPOS_; ++k) g[k] = ar[k];
        } else {
            const float* ar = attn + abase + (size_t)(j * HW_ + i) * SEQ_ + i;
            #pragma unroll
            for (int k = 0; k < NPOS_; ++k) g[k] = ar[k * HW_];
        }
        #pragma unroll
        for (int k = 0; k < NPOS_; ++k)
            g[k] = 1.f / (1.f + __expf(-g[k]));            // sigmoid
        float run = 0.f, pos[NPOS_];
        #pragma unroll
        for (int k = NPOS_ - 1; k >= 0; --k) { run += g[k]; pos[k] = run; }

        const int nrow = (branch == 0) ? (i * HW_ + j) : (j * HW_ + i);
        float* orow = (branch == 0 ? pe_x : pe_y) + ((size_t)bh * SEQ_ + nrow) * NPOS_;
        float ssum = 0.f;
        #pragma unroll
        for (int k = 0; k < NPOS_; ++k) {
            float p  = fminf(pos[k], (float)(NPOS_ - 1));
            float pf = floorf(p);
            int   fi = (int)pf;
            int   ci = (int)ceilf(p);
            float w  = p - pf;
            float lf = li_s[wv][j * NPOS_ + fi];
            float lc = li_s[wv][j * NPOS_ + ci];
            float val = lc * w + lf * (1.f - w);
            orow[k] = val;
            ssum = __builtin_fmaf(val, val, ssum);
        }
        (branch == 0 ? sq_x : sq_y)[(size_t)bh * SEQ_ + nrow] = ssum;
    }
}

// ---------------------------------------------------------------------------
// Stage 2: fused double cdist via f32 WMMA Gram matrices.
// One wave = one 16x16 output tile of one bh matrix.
//   Gram = X X^T accumulated with V_WMMA_F32_16X16X4_F32 (K = 24 = 6 x 4)
//   out  = wx*sqrt(relu(sqx_i+sqx_j-2Gx)) + (1-wx)*sqrt(relu(sqy_i+sqy_j-2Gy))
// A layout (16x4 f32): lane%16 = M row, VGPR0/1 = K = 2*(lane/16) (+1).
// B layout (4x16 f32): lane%16 = N col, same K striping -> for B = X^T the
// load pattern is identical to A with rows from the j-tile.
// ---------------------------------------------------------------------------
__global__ __launch_bounds__(256) void cdist_wmma_stage2(
    const float* __restrict__ pex, const float* __restrict__ pey,
    const float* __restrict__ sqx, const float* __restrict__ sqy,
    const float* __restrict__ wxp, float* __restrict__ out)
{
    const int lane = threadIdx.x & 31;
    const int wv   = threadIdx.x >> 5;
    const int g    = blockIdx.x * 8 + wv;        // wave id in [0, 96*36*36)
    const int bh   = g / (TILES_ * TILES_);
    const int t    = g % (TILES_ * TILES_);
    const int it   = t / TILES_;
    const int jt   = t % TILES_;

    const int m    = lane & 15;
    const int koff = (lane >> 4) << 1;           // 0 or 2
    const int rowA = it * 16 + m;
    const int rowB = jt * 16 + m;

    const float* Xx = pex + (size_t)bh * SEQ_ * NPOS_;
    const float* Xy = pey + (size_t)bh * SEQ_ * NPOS_;

    v8f accX = {};
    v8f accY = {};
    #pragma unroll
    for (int kb = 0; kb < NPOS_; kb += 4) {
        v2f ax = *(const v2f*)(Xx + rowA * NPOS_ + kb + koff);
        v2f bx = *(const v2f*)(Xx + rowB * NPOS_ + kb + koff);
        accX = __builtin_amdgcn_wmma_f32_16x16x4_f32(
            false, ax, false, bx, (short)0, accX, false, false);
        v2f ay = *(const v2f*)(Xy + rowA * NPOS_ + kb + koff);
        v2f by = *(const v2f*)(Xy + rowB * NPOS_ + kb + koff);
        accY = __builtin_amdgcn_wmma_f32_16x16x4_f32(
            false, ay, false, by, (short)0, accY, false, false);
    }

    const float wx = *wxp;
    const float* sX = sqx + (size_t)bh * SEQ_;
    const float* sY = sqy + (size_t)bh * SEQ_;
    const int col = jt * 16 + (lane & 15);
    const float sxc = sX[col];
    const float syc = sY[col];
    const int rbase = it * 16 + ((lane >> 4) << 3);   // upper half-wave: +8
    float* o = out + (size_t)bh * SEQ_ * SEQ_;

    #pragma unroll
    for (int r = 0; r < 8; ++r) {
        const int row = rbase + r;
        float d2x = sX[row] + sxc - 2.0f * accX[r];
        float d2y = sY[row] + syc - 2.0f * accY[r];
        float dx = d2x > 0.f ? __builtin_sqrtf(d2x) : 0.f;
        float dy = d2y > 0.f ? __builtin_sqrtf(d2y) : 0.f;
        o[(size_t)row * SEQ_ + col] = wx * dx + (1.f - wx) * dy;
    }
}

extern "C" void kernel_launch(void* const* d_in, const int* in_sizes, int n_in,
                              void* d_out, int out_size, void* d_ws, size_t ws_size,
                              hipStream_t stream) {
    const float* query  = (const float*)d_in[0];
    const float* attn   = (const float*)d_in[1];
    const float* pemb_x = (const float*)d_in[2];
    const float* pemb_y = (const float*)d_in[3];
    const float* wxp    = (const float*)d_in[4];
    float* out = (float*)d_out;

    // workspace: pe_x, pe_y point sets + squared norms (~10.5 MB)
    const size_t npe = (size_t)BH_ * SEQ_ * NPOS_;   // 1,327,104 floats each
    float* pex = (float*)d_ws;
    float* pey = pex + npe;
    float* sqx = pey + npe;
    float* sqy = sqx + (size_t)BH_ * SEQ_;

    // Stage 1: 2304 units per branch, 4 waves/block, branch = blockIdx.y
    dim3 g1((B_ * HEADS_ * HW_) / 4, 2);             // (576, 2)
    cope_stage1<<<g1, 128, 0, stream>>>(query, attn, pemb_x, pemb_y,
                                        pex, pey, sqx, sqy);

    // Stage 2: one wave per 16x16 tile: 96 * 36 * 36 = 124,416 waves, 8/block
    const int nblocks = (BH_ * TILES_ * TILES_) / 8; // 15,552
    cdist_wmma_stage2<<<nblocks, 256, 0, stream>>>(pex, pey, sqx, sqy, wxp, out);
}